// MpnCovPoolLayer_78271484002416
// MI455X (gfx1250) — compile-verified
//
#include <hip/hip_runtime.h>
#include <hip/hip_bf16.h>

typedef float v2f __attribute__((ext_vector_type(2)));
typedef float v8f __attribute__((ext_vector_type(8)));

#define CDIM   256
#define NSP    4096
#define NBATCH 64

// ---------------- Tensor Data Mover (TDM) tile load: global -> LDS --------------
#if __has_builtin(__builtin_amdgcn_tensor_load_to_lds)
#define HAVE_TDM 1
typedef unsigned int u32x4 __attribute__((ext_vector_type(4)));
typedef int          i32x4 __attribute__((ext_vector_type(4)));
typedef int          i32x8 __attribute__((ext_vector_type(8)));

// Loads a tile of tile_h rows x tile_w fp32 elements starting at gsrc
// (row stride row_stride_elems) into LDS at lds_byte_addr, inserting
// (pad_amount_code+1) dwords of padding after every pad-interval dwords.
// pad_interval_code: 4 -> every 32 dwords, 5 -> every 64 dwords.
__device__ __forceinline__ void tdm_load_tile_f32(
    const float* gsrc, unsigned lds_byte_addr,
    unsigned tile_w, unsigned tile_h,
    unsigned tensor_w, unsigned tensor_h,
    unsigned row_stride_elems,
    unsigned pad_interval_code, unsigned pad_amount_code) {
  unsigned long long ga = (unsigned long long)(size_t)gsrc;
  u32x4 g0;
  g0[0] = 1u;                                   // count=1, user descriptor
  g0[1] = lds_byte_addr;                        // D#.lds_addr (bytes)
  g0[2] = (unsigned)ga;                         // D#.global_addr[31:0]
  g0[3] = (unsigned)((ga >> 32) & 0x01FFFFFFu)  // D#.global_addr[56:32]
        | (2u << 30);                           // D#.type = 2 ("image")
  i32x8 g1;
  g1[0] = (int)((2u << 16)                      // data_size = 4 bytes
        | (1u << 20)                            // pad_enable
        | (pad_interval_code << 22)             // pad_interval
        | (pad_amount_code << 25));             // pad_amount
  g1[1] = (int)((tensor_w & 0xFFFFu) << 16);                                  // tensor_dim0[15:0]
  g1[2] = (int)(((tensor_w >> 16) & 0xFFFFu) | ((tensor_h & 0xFFFFu) << 16)); // tdim0[31:16]|tdim1[15:0]
  g1[3] = (int)(((tensor_h >> 16) & 0xFFFFu) | ((tile_w & 0xFFFFu) << 16));   // tdim1[31:16]|tile_dim0
  g1[4] = (int)(tile_h & 0xFFFFu);              // tile_dim1 (tile_dim2 = 0)
  g1[5] = (int)row_stride_elems;                // tensor_dim0_stride[31:0]
  g1[6] = 0;                                    // stride[47:32]=0, dim1_stride lo=0
  g1[7] = 0;                                    // dim1_stride hi = 0
  i32x4 gz4 = {0, 0, 0, 0};                     // D# group 2 (unused, 2-D tile)
  i32x8 gz8 = {0, 0, 0, 0, 0, 0, 0, 0};         // D# group 3 / ext (unused)
  __builtin_amdgcn_tensor_load_to_lds(g0, g1, gz4, gz4, gz8, 0);
}
#else
#define HAVE_TDM 0
#endif

// ---------------- mean over spatial dim: one block per (b,c) row ----------------
__global__ __launch_bounds__(256) void mean_kernel(const float* __restrict__ x,
                                                   float* __restrict__ means) {
  __shared__ float red[256];
  const int row = blockIdx.x;                 // b*C + c
  const float* xr = x + (size_t)row * NSP;
  float s = 0.f;
  for (int i = threadIdx.x; i < NSP; i += 256) s += xr[i];
  red[threadIdx.x] = s;
  __syncthreads();
  for (int off = 128; off > 0; off >>= 1) {
    if ((int)threadIdx.x < off) red[threadIdx.x] += red[threadIdx.x + off];
    __syncthreads();
  }
  if (threadIdx.x == 0) means[row] = red[0] * (1.0f / NSP);
}

// ---------------- covariance: cov = (X X^T - N m m^T) / (N-1) via f32 WMMA ------
// block tile 128(M) x 64(N); A tile DMA'd by the TDM (padded stride 36), B tile
// manually transposed into LDS. 8 waves (4x2), each wave computes 32x32.
__global__ __launch_bounds__(256) void cov_kernel(const float* __restrict__ x,
                                                  const float* __restrict__ means,
                                                  float* __restrict__ cov) {
  __shared__ float As[128][36];   // [m][k], padded stride 36
  __shared__ float Bs[32][68];    // [k][n], padded stride 68
  const int b    = blockIdx.z;
  const int mb   = blockIdx.x * 128;
  const int nb   = blockIdx.y * 64;
  const float* xb = x + (size_t)b * CDIM * NSP;
  const int tid  = threadIdx.x;
  const int lane = tid & 31;
  const int wave = tid >> 5;
  const int wm   = wave >> 1;     // 0..3 -> M sub-block
  const int wn   = wave & 1;      // 0..1 -> N sub-block
  const int half = lane >> 4;     // K-pair select (ISA A/B f32 layout)
  const int l15  = lane & 15;

  v8f acc[2][2] = {};

  const int bn   = tid >> 2;          // B: channel column 0..63
  const int bk   = (tid & 3) * 8;     // B: 8 k-values each (transposed store)
#if HAVE_TDM
  const unsigned asAddr = (unsigned)(size_t)&As[0][0];
#else
  const int am   = tid >> 1;
  const int acol = (tid & 1) * 16;
#endif

  for (int k0 = 0; k0 < NSP; k0 += 32) {
#if HAVE_TDM
    if (wave == 0)  // one TDM descriptor per chunk, issued by wave 0
      tdm_load_tile_f32(xb + (size_t)mb * NSP + k0, asAddr,
                        32, 128, NSP, CDIM, NSP, /*pad every 32dw*/4, /*+4dw*/3);
#else
    const float4* asrc = (const float4*)(xb + (size_t)(mb + am) * NSP + k0 + acol);
    float4 av0 = asrc[0], av1 = asrc[1], av2 = asrc[2], av3 = asrc[3];
    *(float4*)&As[am][acol + 0]  = av0;
    *(float4*)&As[am][acol + 4]  = av1;
    *(float4*)&As[am][acol + 8]  = av2;
    *(float4*)&As[am][acol + 12] = av3;
#endif
    const float* bsrc = xb + (size_t)(nb + bn) * NSP + k0 + bk;
#pragma unroll
    for (int i = 0; i < 8; ++i) Bs[bk + i][bn] = bsrc[i];  // transpose X -> X^T tile
#if HAVE_TDM
    __builtin_amdgcn_s_wait_tensorcnt(0);   // waves 1..7 trivially pass
#endif
    __syncthreads();

#pragma unroll
    for (int k4 = 0; k4 < 32; k4 += 4) {
      const int ka = k4 + half * 2;
      v2f a0 = *(const v2f*)&As[wm * 32 + l15][ka];
      v2f a1 = *(const v2f*)&As[wm * 32 + 16 + l15][ka];
      v2f b0, b1;
      b0.x = Bs[ka][wn * 32 + l15];       b0.y = Bs[ka + 1][wn * 32 + l15];
      b1.x = Bs[ka][wn * 32 + 16 + l15];  b1.y = Bs[ka + 1][wn * 32 + 16 + l15];
      acc[0][0] = __builtin_amdgcn_wmma_f32_16x16x4_f32(false, a0, false, b0, (short)0, acc[0][0], false, false);
      acc[0][1] = __builtin_amdgcn_wmma_f32_16x16x4_f32(false, a0, false, b1, (short)0, acc[0][1], false, false);
      acc[1][0] = __builtin_amdgcn_wmma_f32_16x16x4_f32(false, a1, false, b0, (short)0, acc[1][0], false, false);
      acc[1][1] = __builtin_amdgcn_wmma_f32_16x16x4_f32(false, a1, false, b1, (short)0, acc[1][1], false, false);
    }
    __syncthreads();
  }

  const float scale = 1.0f / (float)(NSP - 1);
  const float* mrow = means + b * CDIM;
  float* covb = cov + (size_t)b * CDIM * CDIM;
#pragma unroll
  for (int s = 0; s < 2; ++s) {
#pragma unroll
    for (int t = 0; t < 2; ++t) {
      const int rbase = mb + wm * 32 + s * 16 + half * 8;
      const int col   = nb + wn * 32 + t * 16 + l15;
      const float mc = mrow[col];
#pragma unroll
      for (int v = 0; v < 8; ++v) {
        const int r = rbase + v;
        covb[r * CDIM + col] = (acc[s][t][v] - (float)NSP * mrow[r] * mc) * scale;
      }
    }
  }
}

// ---------------- batched 256x256x256 fp32 WMMA GEMM for Newton-Schulz ----------
// mode 0: Out = A@B    mode 1: Out = 1.5*I - 0.5*(A@B)
// TDM double-buffered staging: waves 0/1 issue A/B tile DMAs for the next K-chunk
// while all 8 waves run WMMA on the current one; a single barrier per chunk.
__global__ __launch_bounds__(256) void ns_gemm_kernel(const float* __restrict__ A,
                                                      const float* __restrict__ Bm,
                                                      float* __restrict__ Out,
                                                      int mode) {
#if HAVE_TDM
  __shared__ float As[2][128][36];
  __shared__ float Bs[2][32][68];
#else
  __shared__ float As[1][128][36];
  __shared__ float Bs[1][32][68];
#endif
  const int b    = blockIdx.z;
  const int mb   = blockIdx.x * 128;
  const int nb   = blockIdx.y * 64;
  const float* Ab = A  + (size_t)b * CDIM * CDIM;
  const float* Bb = Bm + (size_t)b * CDIM * CDIM;
  const int tid  = threadIdx.x;
  const int lane = tid & 31;
  const int wave = tid >> 5;
  const int wm   = wave >> 1;
  const int wn   = wave & 1;
  const int half = lane >> 4;
  const int l15  = lane & 15;

  v8f acc[2][2] = {};

#if HAVE_TDM
  const unsigned asAddr0 = (unsigned)(size_t)&As[0][0][0];
  const unsigned asAddr1 = (unsigned)(size_t)&As[1][0][0];
  const unsigned bsAddr0 = (unsigned)(size_t)&Bs[0][0][0];
  const unsigned bsAddr1 = (unsigned)(size_t)&Bs[1][0][0];
  if (wave == 0)
    tdm_load_tile_f32(Ab + (size_t)mb * CDIM, asAddr0, 32, 128, CDIM, CDIM, CDIM, 4, 3);
  else if (wave == 1)
    tdm_load_tile_f32(Bb + nb,                bsAddr0, 64, 32,  CDIM, CDIM, CDIM, 5, 3);

  for (int kc = 0; kc < CDIM / 32; ++kc) {
    const int cur = kc & 1;
    __builtin_amdgcn_s_wait_tensorcnt(0);   // wave 0/1 wait for cur tiles
    __syncthreads();                        // cur visible; cur^1 free (WAR safe)
    if (kc + 1 < CDIM / 32) {
      const int k0n = (kc + 1) * 32;
      if (wave == 0)
        tdm_load_tile_f32(Ab + (size_t)mb * CDIM + k0n, cur ? asAddr0 : asAddr1,
                          32, 128, CDIM, CDIM, CDIM, 4, 3);
      else if (wave == 1)
        tdm_load_tile_f32(Bb + (size_t)k0n * CDIM + nb, cur ? bsAddr0 : bsAddr1,
                          64, 32, CDIM, CDIM, CDIM, 5, 3);
    }
#else
  const int am   = tid >> 1;
  const int acol = (tid & 1) * 16;
  const int bkk  = tid >> 3;
  const int bn8  = (tid & 7) * 8;
  for (int kc = 0; kc < CDIM / 32; ++kc) {
    const int cur = 0;
    const int k0 = kc * 32;
    const float4* asrc = (const float4*)(Ab + (size_t)(mb + am) * CDIM + k0 + acol);
    float4 av0 = asrc[0], av1 = asrc[1], av2 = asrc[2], av3 = asrc[3];
    *(float4*)&As[0][am][acol + 0]  = av0;
    *(float4*)&As[0][am][acol + 4]  = av1;
    *(float4*)&As[0][am][acol + 8]  = av2;
    *(float4*)&As[0][am][acol + 12] = av3;
    const float4* bsrc = (const float4*)(Bb + (size_t)(k0 + bkk) * CDIM + nb + bn8);
    float4 bv0 = bsrc[0], bv1 = bsrc[1];
    *(float4*)&Bs[0][bkk][bn8 + 0] = bv0;
    *(float4*)&Bs[0][bkk][bn8 + 4] = bv1;
    __syncthreads();
#endif

#pragma unroll
    for (int k4 = 0; k4 < 32; k4 += 4) {
      const int ka = k4 + half * 2;
      v2f a0 = *(const v2f*)&As[cur][wm * 32 + l15][ka];
      v2f a1 = *(const v2f*)&As[cur][wm * 32 + 16 + l15][ka];
      v2f b0, b1;
      b0.x = Bs[cur][ka][wn * 32 + l15];       b0.y = Bs[cur][ka + 1][wn * 32 + l15];
      b1.x = Bs[cur][ka][wn * 32 + 16 + l15];  b1.y = Bs[cur][ka + 1][wn * 32 + 16 + l15];
      acc[0][0] = __builtin_amdgcn_wmma_f32_16x16x4_f32(false, a0, false, b0, (short)0, acc[0][0], false, false);
      acc[0][1] = __builtin_amdgcn_wmma_f32_16x16x4_f32(false, a0, false, b1, (short)0, acc[0][1], false, false);
      acc[1][0] = __builtin_amdgcn_wmma_f32_16x16x4_f32(false, a1, false, b0, (short)0, acc[1][0], false, false);
      acc[1][1] = __builtin_amdgcn_wmma_f32_16x16x4_f32(false, a1, false, b1, (short)0, acc[1][1], false, false);
    }
#if !HAVE_TDM
    __syncthreads();
#endif
  }

  float* Ob = Out + (size_t)b * CDIM * CDIM;
#pragma unroll
  for (int s = 0; s < 2; ++s) {
#pragma unroll
    for (int t = 0; t < 2; ++t) {
      const int rbase = mb + wm * 32 + s * 16 + half * 8;
      const int col   = nb + wn * 32 + t * 16 + l15;
#pragma unroll
      for (int v = 0; v < 8; ++v) {
        const int r = rbase + v;
        float val = acc[s][t][v];
        if (mode == 1) val = ((r == col) ? 1.5f : 0.0f) - 0.5f * val;
        Ob[r * CDIM + col] = val;
      }
    }
  }
}

// ---------------- trace per batch ----------------
__global__ __launch_bounds__(256) void trace_kernel(const float* __restrict__ cov,
                                                    float* __restrict__ tr) {
  __shared__ float red[256];
  const int b = blockIdx.x;
  const int c = threadIdx.x;
  red[c] = cov[(size_t)b * CDIM * CDIM + (size_t)c * (CDIM + 1)];
  __syncthreads();
  for (int off = 128; off > 0; off >>= 1) {
    if (c < off) red[c] += red[c + off];
    __syncthreads();
  }
  if (c == 0) tr[b] = red[0];
}

// ---------------- Y0 = cov/trace, Z0 = I ----------------
__global__ __launch_bounds__(256) void init_yz_kernel(const float* __restrict__ cov,
                                                      const float* __restrict__ tr,
                                                      float* __restrict__ Y,
                                                      float* __restrict__ Z) {
  const int idx = blockIdx.x * 256 + threadIdx.x;   // over 64*256*256
  const int b  = idx >> 16;
  const int rc = idx & 65535;
  const float rt = 1.0f / fmaxf(tr[b], 1e-6f);
  Y[idx] = cov[idx] * rt;
  Z[idx] = ((rc >> 8) == (rc & 255)) ? 1.0f : 0.0f;
}

// ---------------- sqrt_mat = sqrt(trace)*Y ; upper-tri gather ----------------
__global__ __launch_bounds__(256) void extract_kernel(const float* __restrict__ Y,
                                                      const float* __restrict__ tr,
                                                      float* __restrict__ out) {
  const int idx = blockIdx.x * 256 + threadIdx.x;   // over 64*256*256
  const int b  = idx >> 16;
  const int rc = idx & 65535;
  const int i = rc >> 8, j = rc & 255;
  if (j < i) return;
  const int p = i * CDIM - (i * (i - 1)) / 2 + (j - i);   // row-major triu index
  const float s = sqrtf(fmaxf(tr[b], 1e-6f));
  out[(size_t)b * ((CDIM * (CDIM + 1)) / 2) + p] = s * Y[idx];
}

extern "C" void kernel_launch(void* const* d_in, const int* in_sizes, int n_in,
                              void* d_out, int out_size, void* d_ws, size_t ws_size,
                              hipStream_t stream) {
  (void)in_sizes; (void)n_in; (void)out_size; (void)ws_size;
  const float* x = (const float*)d_in[0];
  float* out = (float*)d_out;
  char* ws = (char*)d_ws;

  float* means = (float*)ws;                       // 64*256 floats
  float* tr    = (float*)(ws + 64 * 1024);         // 64 floats
  const size_t MATB = (size_t)NBATCH * CDIM * CDIM * sizeof(float);  // 16 MB
  char* mats = ws + (1 << 17);
  float* covT = (float*)(mats + 0 * MATB);         // cov, reused as T
  float* Y    = (float*)(mats + 1 * MATB);
  float* Z    = (float*)(mats + 2 * MATB);
  float* Y2   = (float*)(mats + 3 * MATB);
  float* Z2   = (float*)(mats + 4 * MATB);

  mean_kernel<<<NBATCH * CDIM, 256, 0, stream>>>(x, means);
  cov_kernel<<<dim3(2, 4, NBATCH), 256, 0, stream>>>(x, means, covT);
  trace_kernel<<<NBATCH, 256, 0, stream>>>(covT, tr);
  init_yz_kernel<<<(NBATCH * CDIM * CDIM) / 256, 256, 0, stream>>>(covT, tr, Y, Z);

  // Newton-Schulz: T = 1.5I - 0.5*Z@Y ; Y <- Y@T ; Z <- T@Z   (5 iterations)
  for (int it = 0; it < 5; ++it) {
    ns_gemm_kernel<<<dim3(2, 4, NBATCH), 256, 0, stream>>>(Z, Y, covT, 1);
    ns_gemm_kernel<<<dim3(2, 4, NBATCH), 256, 0, stream>>>(Y, covT, Y2, 0);
    ns_gemm_kernel<<<dim3(2, 4, NBATCH), 256, 0, stream>>>(covT, Z, Z2, 0);
    float* t;
    t = Y; Y = Y2; Y2 = t;
    t = Z; Z = Z2; Z2 = t;
  }

  extract_kernel<<<(NBATCH * CDIM * CDIM) / 256, 256, 0, stream>>>(Y, tr, out);
}